// pointnet_fp_module_87969520157199
// MI455X (gfx1250) — compile-verified
//
#include <hip/hip_runtime.h>
#include <math.h>

#define BN_EPS 1e-3f

typedef __attribute__((ext_vector_type(2))) float v2f;
typedef __attribute__((ext_vector_type(8))) float v8f;
typedef __attribute__((ext_vector_type(4))) unsigned int u32x4;
typedef __attribute__((ext_vector_type(8))) int i32x8;
typedef __attribute__((ext_vector_type(4))) int i32x4;

// Problem dims (fixed by the reference's setup_inputs)
constexpr int B_  = 4;
constexpr int N_  = 16384;
constexpr int M_  = 4096;
constexpr int C1_ = 128;
constexpr int C2_ = 256;
constexpr int K0_ = C2_ + C1_;   // 384 (concat width)
constexpr int H_  = 256;         // hidden
constexpr int CO_ = 128;         // output channels
constexpr int BN_ = B_ * N_;     // 65536 rows

// ---------------------------------------------------------------------------
// Kernel A: 3-NN + inverse-distance interpolate + concat  ->  x0 [BN, 384]
// ---------------------------------------------------------------------------
__global__ __launch_bounds__(256) void knn_interp_concat(
    const float* __restrict__ xyz1, const float* __restrict__ xyz2,
    const float* __restrict__ points1, const float* __restrict__ points2,
    float* __restrict__ x0)
{
    __shared__ float sx[256 * 3];     // xyz2 tile (256 points)
    __shared__ int   sIdx[256][3];
    __shared__ float sW[256][3];

    const int t     = threadIdx.x;
    const int qBase = blockIdx.x * 256;          // N_ % 256 == 0 -> same batch
    const int q     = qBase + t;
    const int b     = qBase / N_;                // uniform across block

    const float px = xyz1[(size_t)q * 3 + 0];
    const float py = xyz1[(size_t)q * 3 + 1];
    const float pz = xyz1[(size_t)q * 3 + 2];

    float d0 = 3.4e38f, d1 = 3.4e38f, d2 = 3.4e38f;
    int   i0 = 0, i1 = 0, i2 = 0;

    const float* x2b = xyz2 + (size_t)b * M_ * 3;
    for (int tile = 0; tile < M_; tile += 256) {
        // cooperatively stage 256 candidate points (768 floats) into LDS
        sx[t        ] = x2b[(size_t)tile * 3 + t        ];
        sx[t + 256  ] = x2b[(size_t)tile * 3 + t + 256  ];
        sx[t + 512  ] = x2b[(size_t)tile * 3 + t + 512  ];
        if (tile + 256 < M_)  // prefetch next tile (global_prefetch_b8)
            __builtin_prefetch(x2b + (size_t)(tile + 256) * 3 + t, 0, 0);
        __syncthreads();
        #pragma unroll 4
        for (int j = 0; j < 256; ++j) {
            const float dx = px - sx[j * 3 + 0];
            const float dy = py - sx[j * 3 + 1];
            const float dz = pz - sx[j * 3 + 2];
            const float d  = fmaf(dx, dx, fmaf(dy, dy, dz * dz));
            const int  gj  = tile + j;
            if (d < d2) {
                if (d < d0)      { d2 = d1; i2 = i1; d1 = d0; i1 = i0; d0 = d; i0 = gj; }
                else if (d < d1) { d2 = d1; i2 = i1; d1 = d;  i1 = gj; }
                else             { d2 = d;  i2 = gj; }
            }
        }
        __syncthreads();
    }

    d0 = fmaxf(d0, 1e-10f); d1 = fmaxf(d1, 1e-10f); d2 = fmaxf(d2, 1e-10f);
    const float iw0 = 1.0f / d0, iw1 = 1.0f / d1, iw2 = 1.0f / d2;
    const float inv = 1.0f / (iw0 + iw1 + iw2);
    sIdx[t][0] = i0;        sIdx[t][1] = i1;        sIdx[t][2] = i2;
    sW[t][0]   = iw0 * inv; sW[t][1]   = iw1 * inv; sW[t][2]   = iw2 * inv;
    __syncthreads();

    // Phase 2: x0 rows [qBase .. qBase+255]: interp (0..255) + concat (256..383)
    const float* p2 = points2 + (size_t)b * M_ * C2_;
    for (int e = t; e < 256 * 96; e += 256) {
        const int lq = e / 96;
        const int c4 = e % 96;
        float4 o;
        if (c4 < 64) {
            const int c  = c4 * 4;
            const int j0 = sIdx[lq][0], j1 = sIdx[lq][1], j2 = sIdx[lq][2];
            const float w0 = sW[lq][0], w1 = sW[lq][1], w2 = sW[lq][2];
            const float4 f0 = *(const float4*)(p2 + (size_t)j0 * C2_ + c);
            const float4 f1 = *(const float4*)(p2 + (size_t)j1 * C2_ + c);
            const float4 f2 = *(const float4*)(p2 + (size_t)j2 * C2_ + c);
            o.x = fmaf(w0, f0.x, fmaf(w1, f1.x, w2 * f2.x));
            o.y = fmaf(w0, f0.y, fmaf(w1, f1.y, w2 * f2.y));
            o.z = fmaf(w0, f0.z, fmaf(w1, f1.z, w2 * f2.z));
            o.w = fmaf(w0, f0.w, fmaf(w1, f1.w, w2 * f2.w));
        } else {
            const int c = (c4 - 64) * 4;
            o = *(const float4*)(points1 + (size_t)(qBase + lq) * C1_ + c);
        }
        *(float4*)(x0 + (size_t)(qBase + lq) * K0_ + c4 * 4) = o;
    }
}

// ---------------------------------------------------------------------------
// TDM helper: async DMA of a 2-D tile (tile1 rows x tile0 f32 cols) from a
// row-major tensor (row length dim0, row stride stride0, in elements) into
// LDS at byte offset lds_off, inserting 4 pad DWORDs after every 32 DWORDs
// (pad_interval=4 -> 32 DW, pad_amount=3 -> 4 DW) => LDS row stride 36 floats.
// Tracked with TENSORcnt.
// ---------------------------------------------------------------------------
__device__ __forceinline__ void tdm_load_2d(
    unsigned lds_off, const float* gptr,
    unsigned dim0, unsigned stride0, unsigned dim1,
    unsigned tile0, unsigned tile1)
{
    const unsigned long long ga = (unsigned long long)(size_t)gptr;
    u32x4 g0;
    g0.x = 1u;                                              // count=1 (valid D#)
    g0.y = lds_off;                                         // lds_addr (bytes)
    g0.z = (unsigned)(ga & 0xffffffffu);                    // global_addr[31:0]
    g0.w = (unsigned)((ga >> 32) & 0x01ffffffu) | (2u << 30); // addr[56:32]|type=2
    i32x8 g1;
    // data_size=2 (4B) | pad_enable | pad_interval=4 (32DW) | pad_amount=3 (4DW)
    g1[0] = (int)((2u << 16) | (1u << 20) | (4u << 22) | (3u << 25));
    g1[1] = (int)((dim0 & 0xffffu) << 16);                  // tensor_dim0 lo16
    g1[2] = (int)(((dim0 >> 16) & 0xffffu) | ((dim1 & 0xffffu) << 16));
    g1[3] = (int)(((dim1 >> 16) & 0xffffu) | ((tile0 & 0xffffu) << 16));
    g1[4] = (int)(tile1 & 0xffffu);                         // tile_dim1, tile_dim2=0
    g1[5] = (int)stride0;                                   // tensor_dim0_stride lo32
    g1[6] = 0;                                              // stride hi / dim1_stride
    g1[7] = 0;
    i32x4 z4 = {0, 0, 0, 0};
#if defined(__clang_major__) && (__clang_major__ >= 23)
    i32x8 z8 = {};
    __builtin_amdgcn_tensor_load_to_lds(g0, g1, z4, z4, z8, 0);
#else
    __builtin_amdgcn_tensor_load_to_lds(g0, g1, z4, z4, 0);
#endif
}

// ---------------------------------------------------------------------------
// WMMA f32 GEMM + fused BN(inference) + ReLU, TDM double-buffered tile feed.
// Block: 256 threads = 8 waves (4x2 grid of 16x16 wave tiles) -> 64x32 tile.
// Wave 0 drives the Tensor Data Mover; everyone syncs on tensorcnt+barrier.
// ---------------------------------------------------------------------------
template <int K, int NOUT>
__global__ __launch_bounds__(256) void gemm_bn_relu(
    const float* __restrict__ X,    // [BN_, K]
    const float* __restrict__ W,    // [K, NOUT]
    const float* __restrict__ gam, const float* __restrict__ bet,
    const float* __restrict__ mu,  const float* __restrict__ var,
    float* __restrict__ Y)          // [BN_, NOUT]
{
    constexpr int KT  = 32;
    constexpr int LDA = 36;                        // padded LDS stride (floats)
    constexpr int ASZ = 64 * LDA;                  // one A buffer (floats)
    constexpr int BSZ = KT * LDA;                  // one B buffer (floats)
    __shared__ float lds_a[2 * ASZ];
    __shared__ float lds_b[2 * BSZ];

    const int t       = threadIdx.x;
    const int lane    = t & 31;
    const int wave    = t >> 5;                    // 0..7
    const int wr      = wave >> 1;                 // 0..3  (row group)
    const int wc      = wave & 1;                  // 0..1  (col group)
    const int laneRow = lane & 15;
    const int half    = lane >> 4;

    const int mBlock = blockIdx.x * 64;
    const int nBlock = blockIdx.y * 32;

    const unsigned aBase = (unsigned)(size_t)(void*)lds_a;  // LDS byte offset
    const unsigned bBase = (unsigned)(size_t)(void*)lds_b;

    v8f acc = {};

    constexpr int nTiles = K / KT;

    // Prologue: DMA tile 0 into buffer 0 (wave 0 only; TDM ignores EXEC).
    if (wave == 0) {
        tdm_load_2d(aBase, X + (size_t)mBlock * K,            K, K,    64, KT, 64);
        tdm_load_2d(bBase, W + nBlock,                        NOUT, NOUT, KT, 32, KT);
    }

    for (int i = 0; i < nTiles; ++i) {
        const int kt  = i * KT;
        const int cur = i & 1;
        const int nxt = cur ^ 1;
        if (wave == 0) {
            if (i + 1 < nTiles) {
                // prefetch next k-tile into the other buffer (its previous
                // consumers finished before the barrier that ended iter i-1)
                tdm_load_2d(aBase + (unsigned)(nxt * ASZ * 4),
                            X + (size_t)mBlock * K + (kt + KT), K, K, 64, KT, 64);
                tdm_load_2d(bBase + (unsigned)(nxt * BSZ * 4),
                            W + (size_t)(kt + KT) * NOUT + nBlock, NOUT, NOUT, KT, 32, KT);
                __builtin_amdgcn_s_wait_tensorcnt(2);   // current pair complete
            } else {
                __builtin_amdgcn_s_wait_tensorcnt(0);
            }
        }
        __syncthreads();                                  // publish buffer `cur`

        const float* la = lds_a + cur * ASZ;
        const float* lb = lds_b + cur * BSZ;
        #pragma unroll
        for (int kk = 0; kk < KT; kk += 4) {
            // A 16x4 f32: lane holds A[M=lane%16][K=2*half], [K=2*half+1]
            const float* ap = la + (wr * 16 + laneRow) * LDA + kk + 2 * half;
            v2f a; a.x = ap[0]; a.y = ap[1];
            // B 4x16 f32: lane holds B[K=2*half][N=lane%16], next K row
            const float* bp = lb + (kk + 2 * half) * LDA + wc * 16 + laneRow;
            v2f bb; bb.x = bp[0]; bb.y = bp[LDA];
            acc = __builtin_amdgcn_wmma_f32_16x16x4_f32(
                false, a, false, bb, (short)0, acc, false, false);
        }
        __syncthreads();                                  // done reading `cur`
    }

    // Fused BN + ReLU epilogue. D layout: VGPR r -> row (8*half + r), col lane%16.
    const int col = nBlock + wc * 16 + laneRow;
    const float s  = gam[col] * rsqrtf(var[col] + BN_EPS);
    const float tt = fmaf(-mu[col], s, bet[col]);
    const int rowBase = mBlock + wr * 16 + half * 8;
    #pragma unroll
    for (int r = 0; r < 8; ++r) {
        float y = fmaf(acc[r], s, tt);
        y = fmaxf(y, 0.0f);
        Y[(size_t)(rowBase + r) * NOUT + col] = y;
    }
}

// ---------------------------------------------------------------------------
// Launch: A -> x0 (ws), GEMM1 -> x1 (ws), GEMM2 -> d_out
// Workspace layout: x0 [BN_*384] f32, then x1 [BN_*256] f32  (~168 MB)
// ---------------------------------------------------------------------------
extern "C" void kernel_launch(void* const* d_in, const int* in_sizes, int n_in,
                              void* d_out, int out_size, void* d_ws, size_t ws_size,
                              hipStream_t stream) {
    (void)in_sizes; (void)n_in; (void)out_size; (void)ws_size;

    const float* xyz1    = (const float*)d_in[0];
    const float* xyz2    = (const float*)d_in[1];
    const float* points1 = (const float*)d_in[2];
    const float* points2 = (const float*)d_in[3];
    const float* w0 = (const float*)d_in[4];
    const float* g0 = (const float*)d_in[5];
    const float* b0 = (const float*)d_in[6];
    const float* m0 = (const float*)d_in[7];
    const float* v0 = (const float*)d_in[8];
    const float* w1 = (const float*)d_in[9];
    const float* g1 = (const float*)d_in[10];
    const float* b1 = (const float*)d_in[11];
    const float* m1 = (const float*)d_in[12];
    const float* v1 = (const float*)d_in[13];

    float* x0  = (float*)d_ws;
    float* x1  = (float*)((char*)d_ws + (size_t)BN_ * K0_ * sizeof(float));
    float* out = (float*)d_out;

    knn_interp_concat<<<dim3(BN_ / 256), dim3(256), 0, stream>>>(
        xyz1, xyz2, points1, points2, x0);

    gemm_bn_relu<K0_, H_><<<dim3(BN_ / 64, H_ / 32), dim3(256), 0, stream>>>(
        x0, w0, g0, b0, m0, v0, x1);

    gemm_bn_relu<H_, CO_><<<dim3(BN_ / 64, CO_ / 32), dim3(256), 0, stream>>>(
        x1, w1, g1, b1, m1, v1, out);
}